// TaskAlignedAssigner_32633161515621
// MI455X (gfx1250) — compile-verified
//
#include <hip/hip_runtime.h>
#include <hip/hip_bf16.h>
#include <stdint.h>

typedef unsigned int u32;
typedef unsigned long long u64;

#define B_   32
#define L_   8400
#define C_   80
#define N_   64
#define TK_  13

// ---------------------------------------------------------------------------
// CDNA5 Tensor Data Mover: 1-D global->LDS DMA (D# built per ISA 08 §8).
// This toolchain exposes the 6-arg builtin:
//   (uint32x4 g0, int32x8 g1, int32x4 g2, int32x4 g3, int32x8 g4, i32 cpol)
// ---------------------------------------------------------------------------
#if defined(__AMDGCN__) && __has_builtin(__builtin_amdgcn_tensor_load_to_lds)
#define HAVE_TDM 1
typedef u32 v4u __attribute__((ext_vector_type(4)));
typedef int v8i __attribute__((ext_vector_type(8)));
typedef int v4i __attribute__((ext_vector_type(4)));

__device__ __forceinline__ void tdm_load_1d(void* lds_dst, const void* gsrc, u32 ndw) {
  u32 lo = (u32)(uintptr_t)lds_dst;            // LDS byte offset (low 32 bits of generic ptr)
  u64 ga = (u64)(uintptr_t)gsrc;
  v4u g0;
  g0[0] = 1u;                                  // count=1, is_restore=0, gather=0
  g0[1] = lo;                                  // lds_addr
  g0[2] = (u32)ga;                             // global_addr[31:0]
  g0[3] = ((u32)(ga >> 32) & 0x01FFFFFFu) | (2u << 30);  // addr[56:32], type=2
  v8i g1;
  g1[0] = (int)(2u << 16);                     // workgroup_mask=0, data_size=2 (4B)
  g1[1] = (int)((ndw & 0xFFFFu) << 16);        // tensor_dim0[15:0]
  g1[2] = (int)(((ndw >> 16) & 0xFFFFu) | (1u << 16)); // tensor_dim0[31:16], tensor_dim1=1
  g1[3] = (int)((ndw & 0xFFFFu) << 16);        // tile_dim0 = ndw
  g1[4] = 1;                                   // tile_dim1=1, tile_dim2=0
  g1[5] = (int)ndw;                            // tensor_dim0_stride[31:0]
  g1[6] = 0;
  g1[7] = 0;
  v4i z4 = {0, 0, 0, 0};
  v8i z8 = {0, 0, 0, 0, 0, 0, 0, 0};
  __builtin_amdgcn_tensor_load_to_lds(g0, g1, z4, z4, z8, 0);
}
__device__ __forceinline__ void tdm_wait() {
#if __has_builtin(__builtin_amdgcn_s_wait_tensorcnt)
  __builtin_amdgcn_s_wait_tensorcnt(0);
#else
  asm volatile("s_wait_tensorcnt 0x0" ::: "memory");
#endif
}
#else
#define HAVE_TDM 0
#endif

// ---------------------------------------------------------------------------
// CIoU, clipped & nan_to_num'd exactly like the reference.
// ---------------------------------------------------------------------------
__device__ __forceinline__ float ciou_clip(
    float g0, float g1, float g2, float g3, float garea, float gcx, float gcy, float gat,
    float p0, float p1, float p2, float p3, float parea, float pcx, float pcy, float pat) {
  float ltx = fmaxf(g0, p0), lty = fmaxf(g1, p1);
  float rbx = fminf(g2, p2), rby = fminf(g3, p3);
  float iw = fmaxf(rbx - ltx, 0.f), ih = fmaxf(rby - lty, 0.f);
  float inter = iw * ih;
  float uni = garea + parea - inter;
  float iou = inter / uni;
  float cwx = fmaxf(fmaxf(g2, p2) - fminf(g0, p0), 0.f);
  float cwy = fmaxf(fmaxf(g3, p3) - fminf(g1, p1), 0.f);
  float diag2 = cwx * cwx + cwy * cwy + 1e-7f;
  float dx = gcx - pcx, dy = gcy - pcy;
  float diou = iou - (dx * dx + dy * dy) / diag2;
  float dv = gat - pat;
  float v = 0.40528473456935109f * dv * dv;     // 4/pi^2
  float alpha = v / (1.0f - iou + v + 1e-7f);
  float c = diou - alpha * v;
  c = isnan(c) ? 0.f : c;
  return fmaxf(c, 0.f);
}

// ---------------------------------------------------------------------------
// Kernel 0: zero the per-(b,gt) atomic-max slots (must run every launch).
// ---------------------------------------------------------------------------
__global__ void kInit(u32* __restrict__ max_am, u32* __restrict__ max_iou) {
  int t = blockIdx.x * 256 + threadIdx.x;
  if (t < B_ * N_) { max_am[t] = 0u; max_iou[t] = 0u; }
}

// ---------------------------------------------------------------------------
// Kernel A: per (b, gt) block. Metric vector (8400 f32) lives in LDS; 13
// stable argmax passes reproduce jax.lax.top_k (ties -> lowest index).
// ---------------------------------------------------------------------------
__global__ __launch_bounds__(256) void kTopK(
    const float* __restrict__ ps, const float4* __restrict__ pbx,
    const float2* __restrict__ ctr, const int* __restrict__ glab,
    const float4* __restrict__ gbox, const float* __restrict__ gpad,
    int* __restrict__ topk) {
  __shared__ float sm[L_];
  __shared__ float rv[256];
  __shared__ int   ri[256];
  int b = blockIdx.x >> 6;
  int i = blockIdx.x & (N_ - 1);
  int tid = threadIdx.x;
  float pd = gpad[b * N_ + i];
  if (pd == 0.f) {                 // padded GT: topk indices are never consulted
    if (tid < TK_) topk[(b * N_ + i) * TK_ + tid] = -1;
    return;
  }
  float4 g = gbox[b * N_ + i];
  float garea = (g.z - g.x) * (g.w - g.y);
  float gcx = (g.x + g.z) * 0.5f, gcy = (g.y + g.w) * 0.5f;
  float gat = atanf((g.z - g.x) / (g.w - g.y));
  int lab = glab[b * N_ + i];
  const float* psb = ps + (size_t)b * L_ * C_ + lab;

  for (int l = tid; l < L_; l += 256) {
    float4 p = pbx[(size_t)b * L_ + l];
    float2 cc = ctr[l];
    float m = fminf(fminf(cc.x - g.x, cc.y - g.y), fminf(g.z - cc.x, g.w - cc.y));
    float metric = 0.f;
    if (m > 1e-9f) {
      float parea = (p.z - p.x) * (p.w - p.y);
      float pcx = (p.x + p.z) * 0.5f, pcy = (p.y + p.w) * 0.5f;
      float pat = atanf((p.z - p.x) / (p.w - p.y));
      float iou = ciou_clip(g.x, g.y, g.z, g.w, garea, gcx, gcy, gat,
                            p.x, p.y, p.z, p.w, parea, pcx, pcy, pat);
      float s = psb[(size_t)l * C_];
      float i2 = iou * iou;
      metric = s * (i2 * i2 * i2);
    }
    sm[l] = metric;
  }
  __syncthreads();

  for (int k = 0; k < TK_; ++k) {
    float bv = -2.f; int bi = 0x7fffffff;
    for (int l = tid; l < L_; l += 256) {
      float v = sm[l];
      if (v > bv) { bv = v; bi = l; }          // ascending l -> keeps lowest idx on ties
    }
    rv[tid] = bv; ri[tid] = bi;
    __syncthreads();
    for (int s = 128; s > 0; s >>= 1) {
      if (tid < s) {
        float ov = rv[tid + s]; int oi = ri[tid + s];
        if (ov > rv[tid] || (ov == rv[tid] && oi < ri[tid])) { rv[tid] = ov; ri[tid] = oi; }
      }
      __syncthreads();
    }
    if (tid == 0) {
      topk[(b * N_ + i) * TK_ + k] = ri[0];
      sm[ri[0]] = -1.0f;
    }
    __syncthreads();
  }
}

// ---------------------------------------------------------------------------
// Shared staging used by kMask/kAssign: GT tables -> LDS via TDM.
// ---------------------------------------------------------------------------
#define STAGE_GT(b_, need_topk_)                                              \
  if (threadIdx.x == 0) {                                                     \
    tdm_ok_stage(sgb, spad, slab, stk, gbox, gpad, glab, topk, (b_), (need_topk_)); \
  }

__device__ __forceinline__ void tdm_ok_stage(
    float4* sgb, float* spad, int* slab, int* stk,
    const float4* gbox, const float* gpad, const int* glab, const int* topk,
    int b, int need_topk) {
#if HAVE_TDM
  tdm_load_1d(sgb,  gbox + (size_t)b * N_, N_ * 4);
  tdm_load_1d(spad, gpad + (size_t)b * N_, N_);
  tdm_load_1d(slab, glab + (size_t)b * N_, N_);
  if (need_topk) tdm_load_1d(stk, topk + (size_t)b * N_ * TK_, N_ * TK_);
  tdm_wait();
#else
  for (int j = 0; j < N_; ++j) {
    sgb[j] = gbox[(size_t)b * N_ + j];
    spad[j] = gpad[(size_t)b * N_ + j];
    slab[j] = glab[(size_t)b * N_ + j];
  }
  if (need_topk)
    for (int j = 0; j < N_ * TK_; ++j) stk[j] = topk[(size_t)b * N_ * TK_ + j];
#endif
}

__device__ __forceinline__ float masked_iou(
    int i, const float4* sgb, const float* spad,
    const float* sarea, const float* sgcx, const float* sgcy, const float* satan,
    float ccx, float ccy,
    float p0, float p1, float p2, float p3, float parea, float pcx, float pcy, float pat,
    bool* valid_out) {
  float pd = spad[i];
  float4 g = sgb[i];
  float m = fminf(fminf(ccx - g.x, ccy - g.y), fminf(g.z - ccx, g.w - ccy));
  bool valid = (pd != 0.f) && (m > 1e-9f);
  *valid_out = valid;
  if (!valid) return 0.f;
  return ciou_clip(g.x, g.y, g.z, g.w, sarea[i], sgcx[i], sgcy[i], satan[i],
                   p0, p1, p2, p3, parea, pcx, pcy, pat);
}

// ---------------------------------------------------------------------------
// Kernel B: per anchor: topk-membership mask over 64 GTs, conflict
// resolution by max-IoU equality (bit-exact via deterministic recompute),
// and atomicMax (float-as-uint, values >= 0) of per-GT normalizers.
// ---------------------------------------------------------------------------
__global__ __launch_bounds__(256) void kMask(
    const float* __restrict__ ps, const float4* __restrict__ pbx,
    const float2* __restrict__ ctr, const int* __restrict__ glab,
    const float4* __restrict__ gbox, const float* __restrict__ gpad,
    const int* __restrict__ topk, u64* __restrict__ bits_ws,
    u32* __restrict__ max_am, u32* __restrict__ max_iou) {
  __shared__ float4 sgb[N_];
  __shared__ float  spad[N_];
  __shared__ int    slab[N_];
  __shared__ int    stk[N_ * TK_];
  __shared__ float  sarea[N_], sgcx[N_], sgcy[N_], satan[N_];
  int b = blockIdx.y;
  STAGE_GT(b, 1);
  __syncthreads();
  if (threadIdx.x < N_) {
    float4 g = sgb[threadIdx.x];
    sarea[threadIdx.x] = (g.z - g.x) * (g.w - g.y);
    sgcx[threadIdx.x] = (g.x + g.z) * 0.5f;
    sgcy[threadIdx.x] = (g.y + g.w) * 0.5f;
    satan[threadIdx.x] = atanf((g.z - g.x) / (g.w - g.y));
  }
  __syncthreads();

  int l = blockIdx.x * 256 + threadIdx.x;
  if (l >= L_) return;
  float4 p = pbx[(size_t)b * L_ + l];
  float2 cc = ctr[l];
  float parea = (p.z - p.x) * (p.w - p.y);
  float pcx = (p.x + p.z) * 0.5f, pcy = (p.y + p.w) * 0.5f;
  float pat = atanf((p.z - p.x) / (p.w - p.y));

  u64 mbits = 0; int sum = 0; float maxIou = 0.f;
  for (int i = 0; i < N_; ++i) {
    bool valid;
    float iou = masked_iou(i, sgb, spad, sarea, sgcx, sgcy, satan,
                           cc.x, cc.y, p.x, p.y, p.z, p.w, parea, pcx, pcy, pat, &valid);
    maxIou = fmaxf(maxIou, iou);
    if (valid) {
      bool intk = false;
#pragma unroll
      for (int k = 0; k < TK_; ++k) intk = intk || (stk[i * TK_ + k] == l);
      if (intk) { mbits |= (1ull << i); ++sum; }
    }
  }

  u64 fbits = 0;
  const float* psrow = ps + ((size_t)b * L_ + l) * C_;
  for (int i = 0; i < N_; ++i) {
    bool cand = (sum > 1) || ((mbits >> i) & 1ull);
    if (!cand) continue;
    bool valid;
    float iou = masked_iou(i, sgb, spad, sarea, sgcx, sgcy, satan,
                           cc.x, cc.y, p.x, p.y, p.z, p.w, parea, pcx, pcy, pat, &valid);
    bool fb = (sum > 1) ? (iou == maxIou) : true;
    if (fb) {
      fbits |= (1ull << i);
      float s = psrow[slab[i]];
      float i2 = iou * iou;
      float met = s * (i2 * i2 * i2);
      atomicMax(&max_am[b * N_ + i], __float_as_uint(met));
      atomicMax(&max_iou[b * N_ + i], __float_as_uint(iou));
    }
  }
  bits_ws[(size_t)b * L_ + l] = fbits;
}

// ---------------------------------------------------------------------------
// Kernel C: final per-anchor assignment + norm computation.
// ---------------------------------------------------------------------------
__global__ __launch_bounds__(256) void kAssign(
    const float* __restrict__ ps, const float4* __restrict__ pbx,
    const float2* __restrict__ ctr, const int* __restrict__ glab,
    const float4* __restrict__ gbox, const float* __restrict__ gpad,
    const int* __restrict__ topk,
    const u64* __restrict__ bits_ws, const u32* __restrict__ max_am,
    const u32* __restrict__ max_iou,
    float* __restrict__ out_lab, float* __restrict__ out_box,
    float* __restrict__ out_fg, int* __restrict__ lab_ws,
    float* __restrict__ norm_ws) {
  __shared__ float4 sgb[N_];
  __shared__ float  spad[N_];
  __shared__ int    slab[N_];
  __shared__ int    stk[1];      // unused here
  __shared__ float  sarea[N_], sgcx[N_], sgcy[N_], satan[N_];
  int b = blockIdx.y;
  STAGE_GT(b, 0);
  __syncthreads();
  if (threadIdx.x < N_) {
    float4 g = sgb[threadIdx.x];
    sarea[threadIdx.x] = (g.z - g.x) * (g.w - g.y);
    sgcx[threadIdx.x] = (g.x + g.z) * 0.5f;
    sgcy[threadIdx.x] = (g.y + g.w) * 0.5f;
    satan[threadIdx.x] = atanf((g.z - g.x) / (g.w - g.y));
  }
  __syncthreads();

  int l = blockIdx.x * 256 + threadIdx.x;
  if (l >= L_) return;
  size_t o = (size_t)b * L_ + l;
  u64 bits = bits_ws[o];
  bool fg = (bits != 0ull);
  int idx = fg ? (__ffsll((long long)bits) - 1) : 0;   // argmax -> first set bit, else 0
  int lab = fg ? slab[idx] : (int)C_;                  // background = 80
  float4 ab = sgb[idx];

  float norm = 0.f;
  if (fg) {
    float4 p = pbx[o];
    float2 cc = ctr[l];
    float parea = (p.z - p.x) * (p.w - p.y);
    float pcx = (p.x + p.z) * 0.5f, pcy = (p.y + p.w) * 0.5f;
    float pat = atanf((p.z - p.x) / (p.w - p.y));
    const float* psrow = ps + o * C_;
    u64 t = bits;
    while (t) {
      int i = __ffsll((long long)t) - 1;
      t &= (t - 1);
      bool valid;
      float iou = masked_iou(i, sgb, spad, sarea, sgcx, sgcy, satan,
                             cc.x, cc.y, p.x, p.y, p.z, p.w, parea, pcx, pcy, pat, &valid);
      float s = psrow[slab[i]];
      float i2 = iou * iou;
      float met = s * (i2 * i2 * i2);
      float scl = met / (__uint_as_float(max_am[b * N_ + i]) + 1e-9f)
                      * __uint_as_float(max_iou[b * N_ + i]);
      norm = fmaxf(norm, scl);
    }
  }
  out_lab[o] = (float)lab;
  out_fg[o] = fg ? 1.f : 0.f;
  ((float4*)out_box)[o] = ab;
  lab_ws[o] = lab;
  norm_ws[o] = norm;
  (void)stk; (void)topk;
}

// ---------------------------------------------------------------------------
// Kernel D: assigned_scores = one_hot(label)[:80] * norm, coalesced (c fastest).
// ---------------------------------------------------------------------------
__global__ __launch_bounds__(256) void kScores(
    const int* __restrict__ lab_ws, const float* __restrict__ norm_ws,
    float* __restrict__ out_sc) {
  long long t = (long long)blockIdx.x * 256 + threadIdx.x;
  const long long total = (long long)B_ * L_ * C_;
  if (t >= total) return;
  int c = (int)(t % C_);
  long long bl = t / C_;
  int lab = lab_ws[bl];
  out_sc[t] = (c == lab) ? norm_ws[bl] : 0.f;
}

// ---------------------------------------------------------------------------
extern "C" void kernel_launch(void* const* d_in, const int* in_sizes, int n_in,
                              void* d_out, int out_size, void* d_ws, size_t ws_size,
                              hipStream_t stream) {
  const float*  ps   = (const float*)d_in[0];
  const float4* pbx  = (const float4*)d_in[1];
  const float2* ctr  = (const float2*)d_in[2];
  const int*    glab = (const int*)d_in[3];
  const float4* gbox = (const float4*)d_in[4];
  const float*  gpad = (const float*)d_in[5];

  char* ws = (char*)d_ws;
  size_t off = 0;
  int* topk = (int*)(ws + off);      off += ((size_t)B_ * N_ * TK_ * 4 + 511) & ~511ull;
  u32* max_am = (u32*)(ws + off);    off += ((size_t)B_ * N_ * 4 + 511) & ~511ull;
  u32* max_iou = (u32*)(ws + off);   off += ((size_t)B_ * N_ * 4 + 511) & ~511ull;
  u64* bits_ws = (u64*)(ws + off);   off += (size_t)B_ * L_ * 8;
  int* lab_ws = (int*)(ws + off);    off += (size_t)B_ * L_ * 4;
  float* norm_ws = (float*)(ws + off); off += (size_t)B_ * L_ * 4;

  float* out_lab = (float*)d_out;
  float* out_box = out_lab + (size_t)B_ * L_;
  float* out_sc  = out_box + (size_t)B_ * L_ * 4;
  float* out_fg  = out_sc  + (size_t)B_ * L_ * C_;

  kInit<<<(B_ * N_ + 255) / 256, 256, 0, stream>>>(max_am, max_iou);
  kTopK<<<B_ * N_, 256, 0, stream>>>(ps, pbx, ctr, glab, gbox, gpad, topk);
  dim3 g2((L_ + 255) / 256, B_);
  kMask<<<g2, 256, 0, stream>>>(ps, pbx, ctr, glab, gbox, gpad, topk,
                                bits_ws, max_am, max_iou);
  kAssign<<<g2, 256, 0, stream>>>(ps, pbx, ctr, glab, gbox, gpad, topk,
                                  bits_ws, max_am, max_iou,
                                  out_lab, out_box, out_fg, lab_ws, norm_ws);
  long long total = (long long)B_ * L_ * C_;
  kScores<<<(unsigned)((total + 255) / 256), 256, 0, stream>>>(lab_ws, norm_ws, out_sc);
}